// GCN_83519934038548
// MI455X (gfx1250) — compile-verified
//
#include <hip/hip_runtime.h>

// ---- problem sizes (fixed by the reference) ----
#define NNODES  50000
#define NEDGES  640000
#define CH      128          // IN_CH == HID == 128
#define NGRAPHS 64

typedef __attribute__((ext_vector_type(2))) float v2f;
typedef __attribute__((ext_vector_type(8))) float v8f;

// ---------------- degree / normalization ----------------
__global__ void k_init_deg(float* __restrict__ deg, int n) {
  int i = blockIdx.x * blockDim.x + threadIdx.x;
  if (i < n) deg[i] = 1.0f;                      // self-loop contributes 1
}

__global__ void k_deg_scatter(const long long* __restrict__ dst,
                              float* __restrict__ deg, int ne) {
  int e = blockIdx.x * blockDim.x + threadIdx.x;
  if (e < ne) atomicAdd(&deg[(int)dst[e]], 1.0f);
}

__global__ void k_rsqrt_inplace(float* __restrict__ deg, int n) {
  int i = blockIdx.x * blockDim.x + threadIdx.x;
  if (i < n) {
    float d = deg[i];
    deg[i] = (d > 0.0f) ? rsqrtf(d) : 0.0f;      // becomes dinv
  }
}

// ---------------- fp32 WMMA GEMM: H = act(X) @ W  (K == N == 128) ------------
// W (128x128, 64 KB) is staged in LDS once per workgroup with an XOR-16 column
// swizzle on rows where (row & 2) != 0 so the two half-waves of a WMMA B-read
// hit disjoint LDS bank groups (no conflicts, no padding needed).
// Each wave computes a full 16x128 output strip: 8 accumulator tiles, 256
// chained V_WMMA_F32_16X16X4_F32 ops, and each X row is read exactly once.
template <bool RELU>
__global__ void k_gemm_wmma_f32(const float* __restrict__ X,
                                const float* __restrict__ W,
                                float* __restrict__ H, int M) {
  __shared__ float ws[CH * CH];                  // 64 KB

  // cooperative staged load of W with bank swizzle (float4 granularity)
  for (int idx = threadIdx.x; idx < CH * (CH / 4); idx += blockDim.x) {
    int r  = idx >> 5;                           // row 0..127
    int c4 = (idx & 31) << 2;                    // col 0,4,...,124
    int cs = c4 ^ ((r & 2) << 3);                // swizzled col
    float4 w = *(const float4*)(W + (size_t)r * CH + c4);
    *(float4*)(ws + (size_t)r * CH + cs) = w;
  }
  __syncthreads();

  int tm = blockIdx.x * (blockDim.x >> 5) + (threadIdx.x >> 5);
  if (tm * 16 >= M) return;                      // whole-wave uniform

  const int lane = threadIdx.x & 31;
  const int half = lane >> 4;                    // K sub-pair select
  const int l15  = lane & 15;

  const float* xrow = X + (size_t)(tm * 16 + l15) * CH;  // A row, read once

  v8f c0 = {}, c1 = {}, c2 = {}, c3 = {}, c4 = {}, c5 = {}, c6 = {}, c7 = {};
  v8f* acc[8] = {&c0, &c1, &c2, &c3, &c4, &c5, &c6, &c7};

  for (int kk = 0; kk < CH; kk += 4) {
    const int ka = kk + half * 2;                // even
    float a0 = xrow[ka];
    float a1 = xrow[ka + 1];
    if (RELU) { a0 = fmaxf(a0, 0.0f); a1 = fmaxf(a1, 0.0f); }
    v2f a; a[0] = a0; a[1] = a1;
    const float* w0 = ws + (size_t)ka * CH;
    const float* w1 = w0 + CH;
    const int swz = (ka & 2) << 3;               // same for rows ka, ka+1
#pragma unroll
    for (int tn = 0; tn < 8; ++tn) {
      int n = (tn * 16 + l15) ^ swz;
      v2f b; b[0] = w0[n]; b[1] = w1[n];
      *acc[tn] = __builtin_amdgcn_wmma_f32_16x16x4_f32(
          false, a, false, b, (short)0, *acc[tn], false, false);
    }
  }

  // C/D layout: VGPR r -> M = tm*16 + r + 8*half, N = tn*16 + l15
  float* outb = H + (size_t)(tm * 16 + half * 8) * CH + l15;
#pragma unroll
  for (int tn = 0; tn < 8; ++tn) {
    v8f c = *acc[tn];
#pragma unroll
    for (int r = 0; r < 8; ++r) outb[(size_t)r * CH + tn * 16] = c[r];
  }
}

// ---------------- A = bias + H * dinv^2 (self-loop term + bias) ----------------
__global__ void k_init_agg(const float* __restrict__ H,
                           const float* __restrict__ dinv,
                           const float* __restrict__ bias,
                           float* __restrict__ A, int n) {
  int idx = blockIdx.x * blockDim.x + threadIdx.x;    // one float4 each
  const int per = CH / 4;
  if (idx >= n * per) return;
  int i  = idx / per;
  int c4 = idx - i * per;
  float di = dinv[i];
  float w  = di * di;
  float4 h = ((const float4*)(H + (size_t)i * CH))[c4];
  float4 b = ((const float4*)bias)[c4];
  float4 o;
  o.x = b.x + h.x * w; o.y = b.y + h.y * w;
  o.z = b.z + h.z * w; o.w = b.w + h.w * w;
  ((float4*)(A + (size_t)i * CH))[c4] = o;
}

// ---------------- edge scatter: A[d] += H[s] * dinv[s]*dinv[d] ----------------
// One wave per edge; lane handles 4 channels (float4 gather, 4 float atomics).
__global__ void k_edge_scatter(const long long* __restrict__ src,
                               const long long* __restrict__ dst,
                               const float* __restrict__ dinv,
                               const float* __restrict__ H,
                               float* __restrict__ A, int ne) {
  int e = blockIdx.x * (blockDim.x >> 5) + (threadIdx.x >> 5);
  if (e >= ne) return;
  int lane = threadIdx.x & 31;
  int s = (int)src[e];
  int d = (int)dst[e];
  float norm = dinv[s] * dinv[d];
  float4 h = ((const float4*)(H + (size_t)s * CH))[lane];
  float* arow = A + (size_t)d * CH + lane * 4;
  atomicAdd(arow + 0, h.x * norm);
  atomicAdd(arow + 1, h.y * norm);
  atomicAdd(arow + 2, h.z * norm);
  atomicAdd(arow + 3, h.w * norm);
}

// ---------------- pooling ----------------
__global__ void k_zero(float* __restrict__ p, int n) {
  int i = blockIdx.x * blockDim.x + threadIdx.x;
  if (i < n) p[i] = 0.0f;
}

__global__ void k_pool_sum(const float* __restrict__ A,
                           const long long* __restrict__ batch,
                           float* __restrict__ sums, int n) {
  int idx = blockIdx.x * blockDim.x + threadIdx.x;
  if (idx >= n * CH) return;
  int i = idx / CH;
  int c = idx - i * CH;
  int g = (int)batch[i];
  atomicAdd(&sums[g * CH + c], fmaxf(A[idx], 0.0f));   // ReLU fused here
}

__global__ void k_pool_cnt(const long long* __restrict__ batch,
                           float* __restrict__ cnt, int n) {
  int i = blockIdx.x * blockDim.x + threadIdx.x;
  if (i < n) atomicAdd(&cnt[(int)batch[i]], 1.0f);
}

__global__ void k_pool_div(const float* __restrict__ sums,
                           const float* __restrict__ cnt,
                           float* __restrict__ g) {
  int idx = blockIdx.x * blockDim.x + threadIdx.x;
  if (idx >= NGRAPHS * CH) return;
  int r = idx / CH;
  g[idx] = sums[idx] / fmaxf(cnt[r], 1.0f);
}

// ---------------- tiny MLP layers (64 rows) ----------------
__global__ void k_mlp(const float* __restrict__ in, const float* __restrict__ W,
                      const float* __restrict__ b, float* __restrict__ out,
                      int rows, int kdim, int ndim, int relu) {
  int idx = blockIdx.x * blockDim.x + threadIdx.x;
  if (idx >= rows * ndim) return;
  int r = idx / ndim;
  int c = idx - r * ndim;
  float acc = b[c];
  for (int k = 0; k < kdim; ++k) acc += in[r * kdim + k] * W[k * ndim + c];
  out[idx] = relu ? fmaxf(acc, 0.0f) : acc;
}

// ---------------- host-side orchestration ----------------
extern "C" void kernel_launch(void* const* d_in, const int* in_sizes, int n_in,
                              void* d_out, int out_size, void* d_ws, size_t ws_size,
                              hipStream_t stream) {
  (void)in_sizes; (void)n_in; (void)out_size; (void)ws_size;

  const float*     x    = (const float*)d_in[0];
  const long long* ei   = (const long long*)d_in[1];      // [2, NEDGES]
  const long long* batch= (const long long*)d_in[2];
  const float* W1  = (const float*)d_in[3];
  const float* b1  = (const float*)d_in[4];
  const float* W2  = (const float*)d_in[5];
  const float* b2  = (const float*)d_in[6];
  const float* Wl1 = (const float*)d_in[7];
  const float* bl1 = (const float*)d_in[8];
  const float* Wl2 = (const float*)d_in[9];
  const float* bl2 = (const float*)d_in[10];
  const float* Wl3 = (const float*)d_in[11];
  const float* bl3 = (const float*)d_in[12];
  float* out = (float*)d_out;

  const long long* src = ei;
  const long long* dst = ei + NEDGES;

  // workspace carve-up (floats)
  float* H    = (float*)d_ws;                 // NNODES*CH
  float* A    = H    + (size_t)NNODES * CH;   // NNODES*CH
  float* dinv = A    + (size_t)NNODES * CH;   // NNODES
  float* sums = dinv + NNODES;                // NGRAPHS*CH
  float* cnt  = sums + NGRAPHS * CH;          // NGRAPHS   (contiguous after sums)
  float* g    = cnt  + NGRAPHS;               // NGRAPHS*CH
  float* t1   = g    + NGRAPHS * CH;          // 64*64
  float* t2   = t1   + 64 * 64;               // 64*32

  const int TB = 256;
  auto cdiv = [](int a, int b) { return (a + b - 1) / b; };

  // degrees -> dinv (in place)
  k_init_deg     <<<cdiv(NNODES, TB), TB, 0, stream>>>(dinv, NNODES);
  k_deg_scatter  <<<cdiv(NEDGES, TB), TB, 0, stream>>>(dst, dinv, NEDGES);
  k_rsqrt_inplace<<<cdiv(NNODES, TB), TB, 0, stream>>>(dinv, NNODES);

  const int rowTiles      = NNODES / 16;                      // 3125
  const int gemmBlocks    = cdiv(rowTiles, TB / 32);          // 391
  const int scatterBlocks = cdiv(NEDGES, TB / 32);            // 80000
  const int aggBlocks     = cdiv(NNODES * (CH / 4), TB);

  // ---- GCN layer 1 ----
  k_gemm_wmma_f32<false><<<gemmBlocks, TB, 0, stream>>>(x, W1, H, NNODES);
  k_init_agg    <<<aggBlocks, TB, 0, stream>>>(H, dinv, b1, A, NNODES);
  k_edge_scatter<<<scatterBlocks, TB, 0, stream>>>(src, dst, dinv, H, A, NEDGES);

  // ---- GCN layer 2 (ReLU of layer 1 fused into GEMM A-load) ----
  k_gemm_wmma_f32<true><<<gemmBlocks, TB, 0, stream>>>(A, W2, H, NNODES);
  k_init_agg    <<<aggBlocks, TB, 0, stream>>>(H, dinv, b2, A, NNODES);
  k_edge_scatter<<<scatterBlocks, TB, 0, stream>>>(src, dst, dinv, H, A, NEDGES);

  // ---- global mean pool (ReLU fused into the read) ----
  k_zero    <<<cdiv(NGRAPHS * CH + NGRAPHS, TB), TB, 0, stream>>>(sums, NGRAPHS * CH + NGRAPHS);
  k_pool_sum<<<cdiv(NNODES * CH, TB), TB, 0, stream>>>(A, batch, sums, NNODES);
  k_pool_cnt<<<cdiv(NNODES, TB), TB, 0, stream>>>(batch, cnt, NNODES);
  k_pool_div<<<cdiv(NGRAPHS * CH, TB), TB, 0, stream>>>(sums, cnt, g);

  // ---- MLP head ----
  k_mlp<<<cdiv(NGRAPHS * 64, TB), TB, 0, stream>>>(g,  Wl1, bl1, t1, NGRAPHS, 128, 64, 1);
  k_mlp<<<cdiv(NGRAPHS * 32, TB), TB, 0, stream>>>(t1, Wl2, bl2, t2, NGRAPHS, 64, 32, 1);
  k_mlp<<<1, NGRAPHS, 0, stream>>>(t2, Wl3, bl3, out, NGRAPHS, 32, 1, 0);
}